// PitchEnergyExtractor_83150566851375
// MI455X (gfx1250) — compile-verified
//
#include <hip/hip_runtime.h>
#include <hip/hip_bf16.h>
#include <stdint.h>

// ---- problem constants (match reference) ----
#define SRATE   16000
#define FRAME   320
#define LAGS    189           // ceil(16000/85)
#define LAG_MIN 5             // ceil(16000/3400)
#define HALFSZ  (LAGS / 2)    // 94
#define WIN_MED 30
#define MED_PAD ((WIN_MED - 1) / 2)   // 14
#define WINLEN  (FRAME + LAGS)        // 509
#define WINPAD  512
#define EPSF    1e-9f
#define NTILES  12            // 12*16 = 192 >= LAGS lag tiles

// ---- CDNA5 async global->LDS path (guarded; falls back to sync staging) ----
#if defined(__has_builtin)
#if __has_builtin(__builtin_amdgcn_global_load_async_to_lds_b128) && \
    __has_builtin(__builtin_amdgcn_global_load_async_to_lds_b32)  && \
    __has_builtin(__builtin_amdgcn_s_wait_asynccnt)
#define USE_ASYNC_LDS 1
#endif
#if __has_builtin(__builtin_amdgcn_wmma_f32_16x16x4_f32)
#define USE_WMMA 1
#endif
#endif
#ifndef USE_ASYNC_LDS
#define USE_ASYNC_LDS 0
#endif
#ifndef USE_WMMA
#define USE_WMMA 0
#endif

typedef int v4i __attribute__((vector_size(4 * sizeof(int))));
typedef __attribute__((address_space(1))) v4i g_v4i_t;
typedef __attribute__((address_space(1))) int g_i32_t;
typedef __attribute__((address_space(3))) v4i l_v4i_t;
typedef __attribute__((address_space(3))) int l_i32_t;
typedef float v2f __attribute__((ext_vector_type(2)));
typedef float v8f __attribute__((ext_vector_type(8)));

// generic->AS1: full VA; generic->AS3: 32-bit truncation (matches LLVM
// addrspacecast lowering on AMDGPU). Integer round-trip also drops const.
__device__ static inline g_v4i_t* g128(const void* p) { return (g_v4i_t*)(uintptr_t)p; }
__device__ static inline g_i32_t* g32 (const void* p) { return (g_i32_t*)(uintptr_t)p; }
__device__ static inline l_v4i_t* l128(void* p) { return (l_v4i_t*)(uint32_t)(uintptr_t)p; }
__device__ static inline l_i32_t* l32 (void* p) { return (l_i32_t*)(uint32_t)(uintptr_t)p; }

// =====================================================================
// Kernel 1: per (b,k) frame -> NCCF over 189 lags, argmax lag, energy
// cross computed on WMMA: D[m,n] += sum_k s1[kb+k] * w[l0+n+kb+k]
// =====================================================================
__global__ __launch_bounds__(256) void pitch_nccf_kernel(
    const float* __restrict__ audio, int T, int K,
    int* __restrict__ idx_out, float* __restrict__ energy_out)
{
  __shared__ __align__(16) float win[WINPAD];
  __shared__ float crossbuf[NTILES * 16];
  __shared__ float e2buf[LAGS];
  __shared__ float nc[LAGS];
  __shared__ float rbuf[256];
  __shared__ float sh_e1;

  const int tid = threadIdx.x;
  const int blk = blockIdx.x;
  const int b   = blk / K;
  const int k   = blk - b * K;
  const float* base = audio + (size_t)b * (size_t)T + (size_t)k * FRAME;
  const int valid = T - k * FRAME;   // floats readable from base (>= FRAME)

  // zero the window: implements the reference's zero padding for tail frames
  if (tid < WINPAD / 4) {
    float4 z; z.x = z.y = z.z = z.w = 0.f;
    *(float4*)&win[tid * 4] = z;
  }
  __syncthreads();

  // stage window into LDS (CDNA5 async global->LDS, ASYNCcnt tracked)
  if (tid < WINPAD / 4) {
    const int c = tid * 4;                       // 16B-aligned (base is too)
#if USE_ASYNC_LDS
    if (c + 4 <= valid) {
      __builtin_amdgcn_global_load_async_to_lds_b128(
          g128(base + c), l128(&win[c]), 0, 0);
    } else if (c < valid) {
      for (int j = 0; j < 4; ++j)
        if (c + j < valid)
          __builtin_amdgcn_global_load_async_to_lds_b32(
              g32(base + c + j), l32(&win[c + j]), 0, 0);
    }
#else
    if (c + 4 <= valid) {
      *(float4*)&win[c] = *(const float4*)(base + c);
    } else {
      for (int j = 0; j < 4; ++j)
        if (c + j < valid) win[c + j] = base[c + j];
    }
#endif
  }
#if USE_ASYNC_LDS
  __builtin_amdgcn_s_wait_asynccnt(0);
#endif
  __syncthreads();

  // e1 = sum(s1^2) over frame (also used for energy)
  float p = 0.f;
  for (int i = tid; i < FRAME; i += 256) { float s = win[i]; p += s * s; }
  rbuf[tid] = p;
  __syncthreads();
  for (int off = 128; off > 0; off >>= 1) {
    if (tid < off) rbuf[tid] += rbuf[tid + off];
    __syncthreads();
  }
  if (tid == 0) sh_e1 = rbuf[0];
  __syncthreads();
  const float e1 = sh_e1;

  // per-lag lagged energy (VALU); cross too when WMMA unavailable
  if (tid < LAGS) {
    const int l = tid + 1;
    float e2 = 0.f;
#if !USE_WMMA
    float cr = 0.f;
#endif
    for (int i = 0; i < FRAME; i += 4) {
      float a0 = win[l + i + 0];
      float a1 = win[l + i + 1];
      float a2 = win[l + i + 2];
      float a3 = win[l + i + 3];
      e2 += a0 * a0 + a1 * a1 + a2 * a2 + a3 * a3;
#if !USE_WMMA
      float4 s = *(const float4*)&win[i];        // uniform: ds broadcast
      cr += s.x * a0 + s.y * a1 + s.z * a2 + s.w * a3;
#endif
    }
    e2buf[tid] = e2;
#if !USE_WMMA
    crossbuf[tid] = cr;
#endif
  }

#if USE_WMMA
  // cross-correlation contraction on the matrix unit.
  // Toeplitz shift lives in B: B[k,n] = w[l0+n+kb+k]; A[m,k] = s1[kb+k]
  // (replicated rows). Every D row accumulates cross[l0+n]; read row 0.
  {
    const int wave = tid >> 5;
    const int lane = tid & 31;
    const int hi   = lane >> 4;          // 0: K=0,1 half | 1: K=2,3 half
    const int q    = hi << 1;
    const int n    = lane & 15;          // lag within tile
    for (int t = wave; t < NTILES; t += 8) {   // wave-uniform: EXEC all-1s
      const int l0 = 1 + (t << 4);
      v8f acc = {0.f, 0.f, 0.f, 0.f, 0.f, 0.f, 0.f, 0.f};
      for (int kb = 0; kb < FRAME; kb += 4) {
        v2f A, Bv;
        A.x  = win[kb + q];                  // s1[kb+q]
        A.y  = win[kb + q + 1];              // s1[kb+q+1]
        Bv.x = win[l0 + n + kb + q];         // w[l0+n+kb+q]      (max 511)
        Bv.y = win[l0 + n + kb + q + 1];
        acc = __builtin_amdgcn_wmma_f32_16x16x4_f32(
            false, A, false, Bv, (short)0, acc, false, false);
      }
      if (hi == 0) {                         // D[0,n] from lanes 0..15
        const int l = l0 + n;
        if (l <= LAGS) crossbuf[l - 1] = acc[0];
      }
    }
  }
#endif
  __syncthreads();

  // nccf = cross / (eps+n1)^2 / (eps+n2)^2
  if (tid < LAGS) {
    const float d1 = EPSF + sqrtf(e1);
    const float d2 = EPSF + sqrtf(e2buf[tid]);
    nc[tid] = crossbuf[tid] / (d1 * d1) / (d2 * d2);
  }
  __syncthreads();

  // first-argmax selection (matches jnp.argmax first-occurrence semantics)
  if (tid == 0) {
    float bv = -3.4e38f; int bj = LAG_MIN;
    for (int j = LAG_MIN; j < LAGS; ++j) {
      float v = nc[j];
      if (v > bv) { bv = v; bj = j; }
    }
    float hv = -3.4e38f; int hj = LAG_MIN;
    for (int j = LAG_MIN; j < HALFSZ; ++j) {
      float v = nc[j];
      if (v > hv) { hv = v; hj = j; }
    }
    int chosen = (hv > 0.99f * bv) ? hj : bj;     // relative idx + LAG_MIN
    idx_out[blk] = chosen + 1;                    // (j-LAG_MIN)+LAG_MIN+1
    energy_out[blk] = e1 * (1.0f / (float)FRAME);
  }
}

// =====================================================================
// Kernel 2: 30-tap sliding lower-median (left replicate pad), f0, voiced
// =====================================================================
__global__ __launch_bounds__(256) void pitch_median_kernel(
    const int* __restrict__ idx,
    float* __restrict__ f0_out, float* __restrict__ voiced_out,
    int K, int NOUT)
{
  __shared__ int sidx[256];
  const int b = blockIdx.x;
  const int t = threadIdx.x;
  if (t < K) sidx[t] = idx[b * K + t];   // one coalesced staging pass
  __syncthreads();
  if (t >= NOUT) return;

  int v[WIN_MED];
#pragma unroll
  for (int m = 0; m < WIN_MED; ++m) {
    int p = t + m - MED_PAD;
    if (p < 0) p = 0;
    v[m] = sidx[p];
  }
  // insertion sort (30 elems), take lower median = v[14]
  for (int a = 1; a < WIN_MED; ++a) {
    int key = v[a];
    int c = a - 1;
    while (c >= 0 && v[c] > key) { v[c + 1] = v[c]; --c; }
    v[c + 1] = key;
  }
  const float med = (float)v[MED_PAD];
  const float f0  = (float)SRATE / (EPSF + med);
  f0_out[b * NOUT + t]     = f0;
  voiced_out[b * NOUT + t] = (f0 > 0.f) ? 1.f : 0.f;
}

// =====================================================================
// Kernel 3: global mean/std over f0 (reference's guarded formulas) + whiten
// =====================================================================
__global__ __launch_bounds__(256) void pitch_whiten_kernel(
    const float* __restrict__ f0, float* __restrict__ wh, int N)
{
  __shared__ float rs[256];
  __shared__ float rc[256];
  __shared__ float sh_mean, sh_std, sh_n;
  const int tid = threadIdx.x;

  float s = 0.f, c = 0.f;
  for (int i = tid; i < N; i += 256) {
    float f = f0[i];
    float v = (f > 0.f) ? 1.f : 0.f;
    s += f * v; c += v;
  }
  rs[tid] = s; rc[tid] = c;
  __syncthreads();
  for (int o = 128; o > 0; o >>= 1) {
    if (tid < o) { rs[tid] += rs[tid + o]; rc[tid] += rc[tid + o]; }
    __syncthreads();
  }
  if (tid == 0) {
    float n = rc[0];
    sh_n = n;
    sh_mean = (n > 0.f) ? rs[0] / fmaxf(n, 1.f) : 0.f;
  }
  __syncthreads();
  const float mean = sh_mean;
  const float n    = sh_n;

  float ss = 0.f;
  for (int i = tid; i < N; i += 256) {
    float f = f0[i];
    float v = (f > 0.f) ? 1.f : 0.f;
    float d = f - mean;
    ss += d * d * v;
  }
  rs[tid] = ss;
  __syncthreads();
  for (int o = 128; o > 0; o >>= 1) {
    if (tid < o) rs[tid] += rs[tid + o];
    __syncthreads();
  }
  if (tid == 0) {
    float var = rs[0] / fmaxf(n - 1.f, 1.f);
    float std = (n > 1.f) ? sqrtf(var) : 1.f;
    if (std == 0.f) std = 1.f;
    sh_std = std;
  }
  __syncthreads();
  const float stdv = sh_std;
  for (int i = tid; i < N; i += 256)
    wh[i] = (f0[i] - mean) / stdv;
}

// =====================================================================
extern "C" void kernel_launch(void* const* d_in, const int* in_sizes, int n_in,
                              void* d_out, int out_size, void* d_ws, size_t ws_size,
                              hipStream_t stream) {
  (void)n_in; (void)out_size; (void)ws_size;

  const float* audio = (const float*)d_in[0];
  const int T = 80000;                         // reference setup
  const int B = in_sizes[0] / T;               // 8
  const int K = (T + FRAME - 1) / FRAME;       // 250 frames
  const int NOUT = K + MED_PAD - WIN_MED + 1;  // 235 smoothed frames

  float* out    = (float*)d_out;
  float* f0     = out;                  // B*NOUT
  float* whiten = out + B * NOUT;       // B*NOUT
  float* voiced = out + 2 * B * NOUT;   // B*NOUT
  float* energy = out + 3 * B * NOUT;   // B*K
  int*   idx    = (int*)d_ws;           // B*K ints

  pitch_nccf_kernel<<<B * K, 256, 0, stream>>>(audio, T, K, idx, energy);
  pitch_median_kernel<<<B, 256, 0, stream>>>(idx, f0, voiced, K, NOUT);
  pitch_whiten_kernel<<<1, 256, 0, stream>>>(f0, whiten, B * NOUT);
}